// TTT_DownProj_Wrapper_79388175499295
// MI455X (gfx1250) — compile-verified
//
#include <hip/hip_runtime.h>

typedef float v2f __attribute__((ext_vector_type(2)));
typedef float v8f __attribute__((ext_vector_type(8)));

namespace {
constexpr int BB   = 2;
constexpr int S    = 4096;
constexpr int KD   = 5632;
constexpr int H    = 2048;
constexpr int E    = 2048;
constexpr int R    = 16;
constexpr int CH   = 256;
constexpr int NCH  = S / CH;    // 16 chunks per batch
constexpr int BN   = BB * NCH;  // 32 total chunks
constexpr float SCALING = 2.0f;          // ALPHA / RANK
constexpr float LR      = 1e-3f * 2.0f;  // TTT_LR * SCALING

constexpr int BM = 128, BNT = 128, KT = 32;
constexpr int LDSP = KT + 4;  // pad to dodge bank conflicts (36 floats = 144B rows)
}

// Generic f32 WMMA GEMM:  C[m,n] (+)= alpha * sum_k A[m,k]*B[n,k]
// VEC=true : requires aK==1, bK==1, K%4==0; float4 global loads + register
//            prefetch pipelining (overlaps global_load_b128 with wmma bursts).
// VEC=false: arbitrary element strides (absorbs all transposes), scalar loads.
template <bool VEC>
__global__ __launch_bounds__(256)
void gemm_f32_wmma(const float* __restrict__ A, const float* __restrict__ Bm,
                   float* __restrict__ Cm,
                   int M, int N, int K,
                   long aRow, long aK, long aBatch,
                   long bRow, long bK, long bBatch,
                   long cRow, long cBatch,
                   float alpha, int accFlag)
{
    __shared__ float As[BM][LDSP];
    __shared__ float Bs[BNT][LDSP];

    const int tid   = threadIdx.x;
    const int lane  = tid & 31;
    const int wave  = tid >> 5;   // 0..7
    const int waveM = wave >> 2;  // 0..1 -> 64-row patch
    const int waveN = wave & 3;   // 0..3 -> 32-col patch
    const int half  = lane >> 4;  // 0/1
    const int l16   = lane & 15;

    const float* Ab = A  + (long)blockIdx.z * aBatch;
    const float* Bb = Bm + (long)blockIdx.z * bBatch;
    float*       Cb = Cm + (long)blockIdx.z * cBatch;

    const int m0 = blockIdx.y * BM;
    const int n0 = blockIdx.x * BNT;

    v8f acc[4][2];
    const v8f vzero = {0.f, 0.f, 0.f, 0.f, 0.f, 0.f, 0.f, 0.f};
#pragma unroll
    for (int mi = 0; mi < 4; ++mi)
#pragma unroll
        for (int ni = 0; ni < 2; ++ni) acc[mi][ni] = vzero;

    // per-thread staging coordinates for the VEC path: 4 float4 per matrix
    // e = i*256 + tid ; row = e/8 ; col4 = (e%8)*4   (8 float4 per 32-wide row)
    float4 pa[4], pb[4];
    const float4 f4z = make_float4(0.f, 0.f, 0.f, 0.f);

    auto prefetch = [&](int k0) {
#pragma unroll
        for (int i = 0; i < 4; ++i) {
            const int e  = i * 256 + tid;
            const int r  = e >> 3;
            const int c4 = (e & 7) * 4;
            const int gk = k0 + c4;
            const int gm = m0 + r;
            pa[i] = (gm < M && gk < K)
                  ? *reinterpret_cast<const float4*>(Ab + (long)gm * aRow + gk) : f4z;
            const int gn = n0 + r;
            pb[i] = (gn < N && gk < K)
                  ? *reinterpret_cast<const float4*>(Bb + (long)gn * bRow + gk) : f4z;
        }
    };

    if (VEC) prefetch(0);

    for (int k0 = 0; k0 < K; k0 += KT) {
        if (VEC) {
            // commit staged registers to LDS
#pragma unroll
            for (int i = 0; i < 4; ++i) {
                const int e  = i * 256 + tid;
                const int r  = e >> 3;
                const int c4 = (e & 7) * 4;
                *reinterpret_cast<float4*>(&As[r][c4]) = pa[i];
                *reinterpret_cast<float4*>(&Bs[r][c4]) = pb[i];
            }
            __syncthreads();
            if (k0 + KT < K) prefetch(k0 + KT);  // loads overlap the wmma burst below
        } else {
            // generic strided staging: 16 scalar elements per thread per matrix
#pragma unroll
            for (int i = 0; i < 16; ++i) {
                const int e  = i * 256 + tid;
                const int r  = e >> 5;
                const int c  = e & 31;
                const int gk = k0 + c;
                const int gm = m0 + r;
                float va = 0.f;
                if (gm < M && gk < K) va = Ab[(long)gm * aRow + (long)gk * aK];
                As[r][c] = va;
                const int gn = n0 + r;
                float vb = 0.f;
                if (gn < N && gk < K) vb = Bb[(long)gn * bRow + (long)gk * bK];
                Bs[r][c] = vb;
            }
            __syncthreads();
        }

#pragma unroll
        for (int ks = 0; ks < KT / 4; ++ks) {
            // f32 WMMA frag layout: lane l holds K = 2*(l/16)+{0,1} for row/col l%16
            const int kb = ks * 4 + half * 2;
            v2f af[4], bf[2];
#pragma unroll
            for (int mi = 0; mi < 4; ++mi)
                af[mi] = *reinterpret_cast<const v2f*>(&As[waveM * 64 + mi * 16 + l16][kb]);
#pragma unroll
            for (int ni = 0; ni < 2; ++ni)
                bf[ni] = *reinterpret_cast<const v2f*>(&Bs[waveN * 32 + ni * 16 + l16][kb]);
#pragma unroll
            for (int mi = 0; mi < 4; ++mi)
#pragma unroll
                for (int ni = 0; ni < 2; ++ni)
                    acc[mi][ni] = __builtin_amdgcn_wmma_f32_16x16x4_f32(
                        false, af[mi], false, bf[ni], (short)0, acc[mi][ni], false, false);
        }
        __syncthreads();
    }

    // C/D layout: lane l, vgpr v <-> (row = v + 8*(l/16), col = l%16)
#pragma unroll
    for (int mi = 0; mi < 4; ++mi) {
#pragma unroll
        for (int ni = 0; ni < 2; ++ni) {
            const int col = n0 + waveN * 32 + ni * 16 + l16;
#pragma unroll
            for (int v = 0; v < 8; ++v) {
                const int row = m0 + waveM * 64 + mi * 16 + half * 8 + v;
                if (row < M && col < N) {
                    const long off = (long)row * cRow + col;
                    float val = alpha * acc[mi][ni][v];
                    if (accFlag) val += Cb[off];
                    Cb[off] = val;
                }
            }
        }
    }
}

// EmbPad[b][j][:] = embed_table[ids[b,j]] for j < S, zeros for j in [S, S+3)
// => conv input row s = contiguous window EmbPad[b][s+1 .. s+3]
__global__ void gather_embed(const float* __restrict__ table,
                             const int* __restrict__ ids,
                             float* __restrict__ embPad)
{
    const long total = (long)BB * (S + 3) * (E / 4);
    const long idx = (long)blockIdx.x * blockDim.x + threadIdx.x;
    if (idx >= total) return;
    const int  e4 = (int)(idx % (E / 4));
    const long bj = idx / (E / 4);
    const int  j  = (int)(bj % (S + 3));
    const int  b  = (int)(bj / (S + 3));
    float4 v = make_float4(0.f, 0.f, 0.f, 0.f);
    if (j < S) {
        const int id = ids[(long)b * S + j];
        v = reinterpret_cast<const float4*>(table + (long)id * E)[e4];
    }
    reinterpret_cast<float4*>(embPad)[idx] = v;
}

// exclusive cumsum over chunks: A_eff[b,n] = init_A - LR * sum_{j<n} dA[b,j]
__global__ void eff_weights_A(const float* __restrict__ initA,
                              const float* __restrict__ dA,
                              float* __restrict__ Aeff)
{
    const int idx = blockIdx.x * blockDim.x + threadIdx.x;
    if (idx >= BB * H * R) return;
    const int b  = idx / (H * R);
    const int hr = idx % (H * R);
    const float base = initA[hr];
    float s = 0.f;
    for (int n = 0; n < NCH; ++n) {
        const long o = (long)(b * NCH + n) * (H * R) + hr;
        Aeff[o] = base - LR * s;
        s += dA[o];
    }
}

__global__ void eff_weights_B(const float* __restrict__ initB,
                              const float* __restrict__ dB,
                              float* __restrict__ Beff)
{
    const int idx = blockIdx.x * blockDim.x + threadIdx.x;
    if (idx >= BB * R * KD) return;
    const int b  = idx / (R * KD);
    const int rk = idx % (R * KD);
    const float base = initB[rk];
    float s = 0.f;
    for (int n = 0; n < NCH; ++n) {
        const long o = (long)(b * NCH + n) * (R * KD) + rk;
        Beff[o] = base - LR * s;
        s += dB[o];
    }
}

static inline dim3 gemm_grid(int M, int N, int Z) {
    return dim3((unsigned)((N + BNT - 1) / BNT), (unsigned)((M + BM - 1) / BM), (unsigned)Z);
}

extern "C" void kernel_launch(void* const* d_in, const int* in_sizes, int n_in,
                              void* d_out, int out_size, void* d_ws, size_t ws_size,
                              hipStream_t stream) {
    (void)in_sizes; (void)n_in; (void)out_size; (void)ws_size;
    const float* x   = (const float*)d_in[0];  // [B,S,K]
    const float* Wb  = (const float*)d_in[1];  // [H,K]
    const float* iA  = (const float*)d_in[2];  // [H,R]
    const float* iB  = (const float*)d_in[3];  // [R,K]
    const float* Wc  = (const float*)d_in[4];  // [E,E,3]
    const float* Wp  = (const float*)d_in[5];  // [H,E]
    const float* tab = (const float*)d_in[6];  // [VOCAB,E]
    const int*   ids = (const int*)d_in[7];    // [B,S]
    float* out = (float*)d_out;                // [B,S,H]
    float* ws  = (float*)d_ws;

    size_t off = 0;
    float* Mbuf   = ws + off; off += (size_t)H * 3 * E;         // fused conv+proj weights [H][3E]
    float* EmbPad = ws + off; off += (size_t)BB * (S + 3) * E;  // gathered+padded embeddings
    float* Vbuf   = ws + off; off += (size_t)BB * S * H;        // V targets
    float* PIbuf  = ws + off; off += (size_t)BN * CH * R;       // proj_in
    float* PEbuf  = ws + off; off += (size_t)BN * CH * R;       // proj_err
    float* dAbuf  = ws + off; off += (size_t)BN * H * R;
    float* dBbuf  = ws + off; off += (size_t)BN * R * KD;
    float* Aeff   = ws + off; off += (size_t)BN * H * R;
    float* Beff   = ws + off; off += (size_t)BN * R * KD;
    float* Midb   = ws + off; off += (size_t)BN * CH * R;

    // 1) gather embeddings with shift padding
    {
        const long total = (long)BB * (S + 3) * (E / 4);
        gather_embed<<<dim3((unsigned)((total + 255) / 256)), 256, 0, stream>>>(tab, ids, EmbPad);
    }

    // 2) Mcat[h][t*E+ein] = sum_eo Wp[h,eo] * Wc[eo,ein,t]   (z = t, strided B reads W_conv directly)
    gemm_f32_wmma<false><<<gemm_grid(H, E, 3), 256, 0, stream>>>(
        Wp, Wc, Mbuf, H, E, E,
        (long)E, 1L, 0L,
        3L, 3L * E, 1L,
        3L * E, (long)E, 1.f, 0);

    // 3) V[b][s][h] = sum_{k<3E} EmbPad[b][s+1..][k] * Mcat[h][k]   (z = batch)
    gemm_f32_wmma<true><<<gemm_grid(S, H, BB), 256, 0, stream>>>(
        EmbPad + E, Mbuf, Vbuf, S, H, 3 * E,
        (long)E, 1L, (long)(S + 3) * E,
        3L * E, 1L, 0L,
        (long)H, (long)S * H, 1.f, 0);

    // 4) base_out = x @ W_base^T  -> d_out
    gemm_f32_wmma<true><<<gemm_grid(BB * S, H, 1), 256, 0, stream>>>(
        x, Wb, out, BB * S, H, KD,
        (long)KD, 1L, 0L,
        (long)KD, 1L, 0L,
        (long)H, 0L, 1.f, 0);

    // 5) proj_in[c,r] = Z @ init_B^T   (z = chunk)
    gemm_f32_wmma<true><<<gemm_grid(CH, R, BN), 256, 0, stream>>>(
        x, iB, PIbuf, CH, R, KD,
        (long)KD, 1L, (long)CH * KD,
        (long)KD, 1L, 0L,
        (long)R, (long)CH * R, 1.f, 0);

    // 6) proj_err[c,r] = V @ init_A  (B strided: B[r][h] = iA[h*R+r])
    gemm_f32_wmma<false><<<gemm_grid(CH, R, BN), 256, 0, stream>>>(
        Vbuf, iA, PEbuf, CH, R, H,
        (long)H, 1L, (long)CH * H,
        1L, (long)R, 0L,
        (long)R, (long)CH * R, 1.f, 0);

    // 7) dA[h,r] = sum_c V[c,h]*proj_in[c,r]  (contraction over c via strides)
    gemm_f32_wmma<false><<<gemm_grid(H, R, BN), 256, 0, stream>>>(
        Vbuf, PIbuf, dAbuf, H, R, CH,
        1L, (long)H, (long)CH * H,
        1L, (long)R, (long)CH * R,
        (long)R, (long)H * R, 1.f, 0);

    // 8) dB[r,k] = sum_c proj_err[c,r]*Z[c,k]
    gemm_f32_wmma<false><<<gemm_grid(R, KD, BN), 256, 0, stream>>>(
        PEbuf, x, dBbuf, R, KD, CH,
        1L, (long)R, (long)CH * R,
        1L, (long)KD, (long)CH * KD,
        (long)KD, (long)R * KD, 1.f, 0);

    // 9) exclusive cumsum -> effective fast weights
    eff_weights_A<<<dim3((BB * H * R + 255) / 256), 256, 0, stream>>>(iA, dAbuf, Aeff);
    eff_weights_B<<<dim3((BB * R * KD + 255) / 256), 256, 0, stream>>>(iB, dBbuf, Beff);

    // 10) mid[c,r] = Z @ B_eff^T
    gemm_f32_wmma<true><<<gemm_grid(CH, R, BN), 256, 0, stream>>>(
        x, Beff, Midb, CH, R, KD,
        (long)KD, 1L, (long)CH * KD,
        (long)KD, 1L, (long)R * KD,
        (long)R, (long)CH * R, 1.f, 0);

    // 11) out += SCALING * mid @ A_eff^T   (K=16, unit strides, K%4==0)
    gemm_f32_wmma<true><<<gemm_grid(CH, H, BN), 256, 0, stream>>>(
        Midb, Aeff, out, CH, H, R,
        (long)R, 1L, (long)CH * R,
        (long)R, 1L, (long)H * R,
        (long)H, (long)CH * H, SCALING, 1);
}